// UNNBiaffineScorer_56779467653673
// MI455X (gfx1250) — compile-verified
//
#include <hip/hip_runtime.h>
#include <math.h>

// Problem constants (from reference)
#define B_    32
#define N_    512
#define DIN   256
#define DH    128
#define NITER 10
#define NM1   511      // N-1 (Laplacian size)
#define LDM   512      // padded leading dim for Lhat / Binv

typedef float v2f __attribute__((ext_vector_type(2)));
typedef float v8f __attribute__((ext_vector_type(8)));
typedef int   v2i __attribute__((ext_vector_type(2)));

// ---------------------------------------------------------------------------
// CDNA5 async global->LDS copy support (guarded; falls back to direct loads)
// Prototypes (from hipcc diagnostics): b32 takes AS1 int* / AS3 int*,
// b64 takes AS1 v2i* / AS3 v2i*; trailing (imm offset, imm cpol).
// ---------------------------------------------------------------------------
#if __has_builtin(__builtin_amdgcn_global_load_async_to_lds_b32) && \
    __has_builtin(__builtin_amdgcn_global_load_async_to_lds_b64) && \
    __has_builtin(__builtin_amdgcn_s_wait_asynccnt)
#define HAVE_ASYNC 1
#else
#define HAVE_ASYNC 0
#endif

#if HAVE_ASYNC
#define GLOBAL_AS __attribute__((address_space(1)))
#define LDS_AS    __attribute__((address_space(3)))
__device__ __forceinline__ void async_cp_b32(const float* g, float* l) {
  __builtin_amdgcn_global_load_async_to_lds_b32(
      (GLOBAL_AS int*)(__UINTPTR_TYPE__)g, (LDS_AS int*)(int*)l, 0, 0);
}
__device__ __forceinline__ void async_cp_b64(const float* g, float* l) {
  __builtin_amdgcn_global_load_async_to_lds_b64(
      (GLOBAL_AS v2i*)(__UINTPTR_TYPE__)g, (LDS_AS v2i*)(v2i*)l, 0, 0);
}
#endif

// LDS B-slab geometry: 32(K) x 64(N) staged as [n][k], stride 34 floats
// (only 16 consecutive n-rows are read per fragment -> conflict-free;
//  stride*4 % 8 == 0 so v2f reads are 8-byte aligned).
#define BSTRIDE 34
#define BSLAB   (64 * BSTRIDE)

// ---------------------------------------------------------------------------
// FP32 WMMA: D(16x16) = A(16x4) * B(4x16) + C   -- v_wmma_f32_16x16x4_f32
// A frag: lanes 0-15 rows M=lane (K=0,1), lanes 16-31 rows M=lane-16 (K=2,3)
// B frag: lanes 0-15 cols N=lane (K=0,1), lanes 16-31 cols N=lane-16 (K=2,3)
// C/D   : vgpr v, lanes 0-15 -> (M=v, N=lane); lanes 16-31 -> (M=v+8, N=lane-16)
// ---------------------------------------------------------------------------
__device__ __forceinline__ v8f wmma_f32(v2f a, v2f b, v8f c) {
  return __builtin_amdgcn_wmma_f32_16x16x4_f32(
      /*neg_a=*/false, a, /*neg_b=*/false, b,
      /*c_mod=*/(short)0, c, /*reuse_a=*/false, /*reuse_b=*/false);
}

template<bool BIAS, bool CADD, bool RAW, bool RELU>
__device__ __forceinline__ void store_tile(const v8f& acc, int row0, int col,
                                           int ldc, const float* __restrict__ Cb,
                                           float* __restrict__ Rb,
                                           float* __restrict__ Db,
                                           float bv, int half)
{
#pragma unroll
  for (int v = 0; v < 8; ++v) {
    const int row = row0 + v + half * 8;
    const long idx = (long)row * ldc + col;
    float x = acc[v];
    if (BIAS) x += bv;
    if (CADD) x += Cb[idx];
    if (RAW)  Rb[idx] = x;
    Db[idx] = RELU ? fmaxf(x, 0.0f) : x;
  }
}

// ---------------------------------------------------------------------------
// Batched fp32 WMMA GEMM, compile-time specialized:
//   D[b] = op(A[b]) * op(B[b]) (+bias)(+Cadd) [ReLU] [also store pre-act]
// 256 threads = 8 waves (4 down M x 2 across N); each wave owns a 32x32 tile.
// The 32K x 64N B slab is staged global->LDS with async copies, double
// buffered against the WMMA inner loop (8 unrolled K=4 steps -> 32 WMMAs).
// A fragments are read directly from global (b64, only 2-way shared).
// All shapes are exact multiples of the 128x64 block tile; no bounds checks.
// ---------------------------------------------------------------------------
template<int OPA, int OPB, bool BIAS, bool CADD, bool RAW, bool RELU>
__global__ __launch_bounds__(256)
void gemm_wmma(const float* __restrict__ Ag, const float* __restrict__ Bg,
               const float* __restrict__ Cadd, const float* __restrict__ bias,
               float* __restrict__ Dg, float* __restrict__ Draw,
               int K, int lda, int ldb, int ldc,
               long sA, long sB, long sC)
{
  const int b = blockIdx.z;
  const float* __restrict__ Ab = Ag + (long)b * sA;
  const float* __restrict__ Bb = Bg + (long)b * sB;
  float* __restrict__ Db = Dg + (long)b * sC;
  const float* __restrict__ Cb = CADD ? (Cadd + (long)b * sC) : nullptr;
  float* __restrict__ Rb = RAW ? (Draw + (long)b * sC) : nullptr;

  const int tid  = threadIdx.x;
  const int lane = tid & 31;
  const int wave = tid >> 5;
  const int wm = wave & 3;                 // 4 waves down M  (4*32 = 128)
  const int wn = wave >> 2;                // 2 waves across N (2*32 = 64)
  const int r0 = blockIdx.x * 128 + wm * 32;
  const int c0g = blockIdx.y * 64;         // block column base
  const int c0 = c0g + wn * 32;
  const int half = lane >> 4;              // selects K pair within fragment
  const int lr   = lane & 15;

  const int arow0 = r0 + lr,      arow1 = r0 + 16 + lr;
  const int bcol0 = c0 + lr,      bcol1 = c0 + 16 + lr;

  // A fragment base pointers (pointer-bump; no muls in the K loop).
  const float *pa0, *pa1;
  if (OPA == 0) {                          // A row-major [M x K]
    pa0 = Ab + (long)arow0 * lda + 2 * half;
    pa1 = Ab + (long)arow1 * lda + 2 * half;
  } else {                                 // use A^T of stored [K x M]
    pa0 = Ab + (long)(2 * half) * lda + arow0;
    pa1 = Ab + (long)(2 * half) * lda + arow1;
  }

  v8f acc00 = {}, acc01 = {}, acc10 = {}, acc11 = {};

#if HAVE_ASYNC
  // ---- async double-buffered LDS staging of the B slab ----
  __shared__ float Blds[2][BSLAB];
  const int sn = tid >> 2;                 // 0..63 : destination LDS row (N)
  const int sk = (tid & 3) * 8;            // 0..24 : K sub-block of 8
  // per-thread async instruction count per stage (== ASYNCcnt delta)
  //   OPB==0: 8 x b32 transposing gather;  OPB==1: 4 x b64 straight copy
  #define STAGE_B(bufi, kbase)                                               \
    do {                                                                     \
      float* dst = &Blds[bufi][sn * BSTRIDE + sk];                           \
      if (OPB == 0) {                                                        \
        const float* src = Bb + (long)((kbase) + sk) * ldb + c0g + sn;       \
        _Pragma("unroll")                                                    \
        for (int i = 0; i < 8; ++i) async_cp_b32(src + (long)i * ldb, dst + i); \
      } else {                                                               \
        const float* src = Bb + (long)(c0g + sn) * ldb + (kbase) + sk;       \
        _Pragma("unroll")                                                    \
        for (int i = 0; i < 4; ++i) async_cp_b64(src + i * 2, dst + i * 2);  \
      }                                                                      \
    } while (0)

  STAGE_B(0, 0);
  int buf = 0;
  const float* bl0base = &Blds[0][(wn * 32 + lr) * BSTRIDE + 2 * half];
  const float* bl1base = bl0base + 16 * BSTRIDE;

  for (int k0 = 0; k0 < K; k0 += 32) {
    if (k0 + 32 < K) {
      STAGE_B(buf ^ 1, k0 + 32);                       // prefetch next slab
      if (OPB == 0) __builtin_amdgcn_s_wait_asynccnt(8);  // current slab done
      else          __builtin_amdgcn_s_wait_asynccnt(4);
      if (OPA == 0) { __builtin_prefetch(pa0 + k0 + 32, 0, 3);
                      __builtin_prefetch(pa1 + k0 + 32, 0, 3); }
    } else {
      __builtin_amdgcn_s_wait_asynccnt(0);
    }
    __syncthreads();                                   // slab visible to all

    const float* bl0 = bl0base + buf * BSLAB;
    const float* bl1 = bl1base + buf * BSLAB;
#pragma unroll
    for (int ku = 0; ku < 32; ku += 4) {
      const int k = k0 + ku;
      v2f a0, a1, bf0, bf1;
      if (OPA == 0) {
        a0 = *(const v2f*)(pa0 + k);
        a1 = *(const v2f*)(pa1 + k);
      } else {
        a0.x = pa0[(long)k * lda];  a0.y = pa0[(long)(k + 1) * lda];
        a1.x = pa1[(long)k * lda];  a1.y = pa1[(long)(k + 1) * lda];
      }
      bf0 = *(const v2f*)(bl0 + ku);                   // ds_load_b64
      bf1 = *(const v2f*)(bl1 + ku);
      acc00 = wmma_f32(a0, bf0, acc00);
      acc01 = wmma_f32(a0, bf1, acc01);
      acc10 = wmma_f32(a1, bf0, acc10);
      acc11 = wmma_f32(a1, bf1, acc11);
    }
    __syncthreads();                                   // all reads done
    buf ^= 1;
  }
  #undef STAGE_B
#else
  // ---- fallback: direct global loads for B fragments ----
  const float *pb0, *pb1;
  if (OPB == 0) {                          // B row-major [K x N]
    pb0 = Bb + (long)(2 * half) * ldb + bcol0;
    pb1 = Bb + (long)(2 * half) * ldb + bcol1;
  } else {                                 // use B^T of stored [N x K]
    pb0 = Bb + (long)bcol0 * ldb + 2 * half;
    pb1 = Bb + (long)bcol1 * ldb + 2 * half;
  }
  for (int k0 = 0; k0 < K; k0 += 32) {
    if (k0 + 32 < K && OPA == 0) {
      __builtin_prefetch(pa0 + k0 + 32, 0, 3);
      __builtin_prefetch(pa1 + k0 + 32, 0, 3);
    }
#pragma unroll
    for (int ku = 0; ku < 32; ku += 4) {
      const int k = k0 + ku;
      v2f a0, a1, bf0, bf1;
      if (OPA == 0) {
        a0 = *(const v2f*)(pa0 + k);
        a1 = *(const v2f*)(pa1 + k);
      } else {
        a0.x = pa0[(long)k * lda];  a0.y = pa0[(long)(k + 1) * lda];
        a1.x = pa1[(long)k * lda];  a1.y = pa1[(long)(k + 1) * lda];
      }
      if (OPB == 0) {
        bf0.x = pb0[(long)k * ldb]; bf0.y = pb0[(long)(k + 1) * ldb];
        bf1.x = pb1[(long)k * ldb]; bf1.y = pb1[(long)(k + 1) * ldb];
      } else {
        bf0 = *(const v2f*)(pb0 + k);
        bf1 = *(const v2f*)(pb1 + k);
      }
      acc00 = wmma_f32(a0, bf0, acc00);
      acc01 = wmma_f32(a0, bf1, acc01);
      acc10 = wmma_f32(a1, bf0, acc10);
      acc11 = wmma_f32(a1, bf1, acc11);
    }
  }
#endif

  const float bv0 = BIAS ? bias[bcol0] : 0.0f;
  const float bv1 = BIAS ? bias[bcol1] : 0.0f;
  store_tile<BIAS, CADD, RAW, RELU>(acc00, r0,      bcol0, ldc, Cb, Rb, Db, bv0, half);
  store_tile<BIAS, CADD, RAW, RELU>(acc01, r0,      bcol1, ldc, Cb, Rb, Db, bv1, half);
  store_tile<BIAS, CADD, RAW, RELU>(acc10, r0 + 16, bcol0, ldc, Cb, Rb, Db, bv0, half);
  store_tile<BIAS, CADD, RAW, RELU>(acc11, r0 + 16, bcol1, ldc, Cb, Rb, Db, bv1, half);
}

// ---------------------------------------------------------------------------
// c[b] = max over pair-masked scores (one workgroup per batch)
// ---------------------------------------------------------------------------
__global__ __launch_bounds__(1024)
void cmax_kernel(const float* __restrict__ scores, const int* __restrict__ lengths,
                 float* __restrict__ cmax)
{
  const int b = blockIdx.x;
  const int len = lengths[b];
  const float* __restrict__ S = scores + (long)b * N_ * N_;
  float mx = -INFINITY;
  for (int idx = threadIdx.x; idx < N_ * N_; idx += 1024) {
    const int h = idx >> 9, m = idx & (N_ - 1);
    const bool pair = (h <= len) && (m >= 1) && (m <= len) && (h != m);
    if (pair) mx = fmaxf(mx, S[idx]);
  }
  __shared__ float red[1024];
  red[threadIdx.x] = mx;
  __syncthreads();
  for (int s = 512; s > 0; s >>= 1) {
    if (threadIdx.x < s) red[threadIdx.x] = fmaxf(red[threadIdx.x], red[threadIdx.x + s]);
    __syncthreads();
  }
  if (threadIdx.x == 0) cmax[b] = red[0];
}

// ---------------------------------------------------------------------------
// A[b,h,m] = pair ? exp(scores - c[b]) : 0   (A aliases the d_out Y region)
// ---------------------------------------------------------------------------
__global__ __launch_bounds__(1024)
void expA_kernel(const float* __restrict__ scores, const int* __restrict__ lengths,
                 const float* __restrict__ cmax, float* __restrict__ A)
{
  const long total = (long)B_ * N_ * N_;
  for (long idx = (long)blockIdx.x * blockDim.x + threadIdx.x; idx < total;
       idx += (long)gridDim.x * blockDim.x) {
    const int b   = (int)(idx >> 18);          // N_*N_ == 2^18
    const int rem = (int)(idx & (N_ * N_ - 1));
    const int h = rem >> 9, m = rem & (N_ - 1);
    const int len = lengths[b];
    const bool pair = (h <= len) && (m >= 1) && (m <= len) && (h != m);
    A[idx] = pair ? expf(scores[idx] - cmax[b]) : 0.0f;
  }
}

// ---------------------------------------------------------------------------
// colsum[b][m] = sum_h A[b,h,m]   (diagonal of the Laplacian: r + colsum(At))
// ---------------------------------------------------------------------------
__global__ __launch_bounds__(512)
void colsum_kernel(const float* __restrict__ A, float* __restrict__ colsum)
{
  const int b = blockIdx.x;
  const int m = threadIdx.x;
  const float* __restrict__ Ab = A + (long)b * N_ * N_;
  float s = 0.0f;
  for (int h = 0; h < N_; ++h) s += Ab[h * N_ + m];
  colsum[b * N_ + m] = s;
}

// ---------------------------------------------------------------------------
// Build Lhat [NM1 x NM1] (leading dim LDM):
//   row 0      : r[j]            = A[0, j+1]
//   i==j (i>0) : colsum[j+1] + (tok ? 0 : 1)
//   else       : -A[i+1, j+1]
// ---------------------------------------------------------------------------
__global__ __launch_bounds__(256)
void lhat_kernel(const float* __restrict__ A, const float* __restrict__ colsum,
                 const int* __restrict__ lengths, float* __restrict__ Lhat)
{
  const int b = blockIdx.z;
  const float* __restrict__ Ab = A + (long)b * N_ * N_;
  float* __restrict__ Lb = Lhat + (long)b * NM1 * LDM;
  const int len = lengths[b];
  for (int idx = blockIdx.x * blockDim.x + threadIdx.x; idx < NM1 * NM1;
       idx += gridDim.x * blockDim.x) {
    const int i = idx / NM1, j = idx % NM1;
    float v;
    if (i == 0)      v = Ab[(j + 1)];                                   // row 0 = r
    else if (i == j) v = colsum[b * N_ + (j + 1)] + (((j + 1) <= len) ? 0.0f : 1.0f);
    else             v = -Ab[(i + 1) * N_ + (j + 1)];
    Lb[i * LDM + j] = v;
  }
}

// ---------------------------------------------------------------------------
// In-place LU (no pivoting; matrix-tree Laplacian) — one 1024-thread WG/batch.
// Also accumulates logdet = sum log|U_kk|.
// ---------------------------------------------------------------------------
__global__ __launch_bounds__(1024)
void lu_kernel(float* __restrict__ Lhat, float* __restrict__ logdet)
{
  const int b = blockIdx.x;
  float* __restrict__ Mx = Lhat + (long)b * NM1 * LDM;
  float ldet = 0.0f;
  for (int k = 0; k < NM1; ++k) {
    __syncthreads();
    const float piv = Mx[k * LDM + k];
    if (threadIdx.x == 0) ldet += logf(fabsf(piv));
    const float inv = 1.0f / piv;
    for (int i = k + 1 + threadIdx.x; i < NM1; i += 1024)
      Mx[i * LDM + k] *= inv;                 // L factors in place
    __syncthreads();
    const int rows = NM1 - k - 1;
    if (rows <= 0) break;
    const int total = rows * rows;
    const float* __restrict__ Urow = Mx + k * LDM;
    for (int idx = threadIdx.x; idx < total; idx += 1024) {
      const int i = k + 1 + idx / rows;
      const int j = k + 1 + idx % rows;
      Mx[i * LDM + j] -= Mx[i * LDM + k] * Urow[j];   // trailing update (j coalesced)
    }
  }
  if (threadIdx.x == 0) logdet[b] = ldet;
}

// ---------------------------------------------------------------------------
// Binv = Lhat^{-1} via per-column forward/backward substitution.
// Columns are independent -> one thread per column, no barriers.
// ---------------------------------------------------------------------------
__global__ __launch_bounds__(512)
void inv_kernel(const float* __restrict__ Lhat, float* __restrict__ Binv)
{
  const int b = blockIdx.x;
  const float* __restrict__ Mx = Lhat + (long)b * NM1 * LDM;
  float* __restrict__ X = Binv + (long)b * NM1 * LDM;
  const int c = threadIdx.x;
  if (c >= NM1) return;
  // forward: L y = e_c (unit diagonal)
  for (int i = 0; i < NM1; ++i) {
    float s = (i == c) ? 1.0f : 0.0f;
    const float* __restrict__ Li = Mx + i * LDM;
    for (int j = 0; j < i; ++j) s -= Li[j] * X[j * LDM + c];
    X[i * LDM + c] = s;
  }
  // backward: U x = y
  for (int i = NM1 - 1; i >= 0; --i) {
    float s = X[i * LDM + c];
    const float* __restrict__ Ui = Mx + i * LDM;
    for (int j = i + 1; j < NM1; ++j) s -= Ui[j] * X[j * LDM + c];
    X[i * LDM + c] = s / Ui[i];
  }
}

// ---------------------------------------------------------------------------
// Marginals: Y = A * w(Binv)  (in place over A), plus dot[b] = sum(Y*scores).
//   h==0: w = Binv[j,0] + (j>=1)*Binv[j,j]          (j = m-1)
//   h>=1: w = (j>=1)*Binv[j,j] - (i>=1)*Binv[j,i]   (i = h-1)
// ---------------------------------------------------------------------------
__global__ __launch_bounds__(1024)
void marg_kernel(float* __restrict__ A, const float* __restrict__ scores,
                 const float* __restrict__ Binv, float* __restrict__ dotYs)
{
  const int b = blockIdx.x;
  float* __restrict__ Ab = A + (long)b * N_ * N_;
  const float* __restrict__ Sb = scores + (long)b * N_ * N_;
  const float* __restrict__ X = Binv + (long)b * NM1 * LDM;
  float dot = 0.0f;
  for (int idx = threadIdx.x; idx < N_ * N_; idx += 1024) {
    const int h = idx >> 9, m = idx & (N_ - 1);
    const float a = Ab[idx];
    float y = 0.0f;
    if (a != 0.0f && m >= 1) {
      const int j = m - 1;
      const float diag = (j >= 1) ? X[j * LDM + j] : 0.0f;
      float w;
      if (h == 0) {
        w = X[j * LDM + 0] + diag;
      } else {
        const int i = h - 1;
        w = diag - ((i >= 1) ? X[j * LDM + i] : 0.0f);
      }
      y = a * w;
    }
    Ab[idx] = y;
    dot += y * Sb[idx];
  }
  __shared__ float red[1024];
  red[threadIdx.x] = dot;
  __syncthreads();
  for (int s = 512; s > 0; s >>= 1) {
    if (threadIdx.x < s) red[threadIdx.x] += red[threadIdx.x + s];
    __syncthreads();
  }
  if (threadIdx.x == 0) dotYs[b] = red[0];
}

// ---------------------------------------------------------------------------
// logZ[b] = logdet + len*c ; entr[b] = logZ - sum(Y*scores)
// ---------------------------------------------------------------------------
__global__ void finalize_kernel(const float* __restrict__ logdet,
                                const float* __restrict__ cmax,
                                const int* __restrict__ lengths,
                                const float* __restrict__ dotYs,
                                float* __restrict__ logZ, float* __restrict__ entr)
{
  const int b = threadIdx.x;
  if (b < B_) {
    const float lz = logdet[b] + (float)lengths[b] * cmax[b];
    logZ[b] = lz;
    entr[b] = lz - dotYs[b];
  }
}

// ---------------------------------------------------------------------------
// Host-side orchestration
// ---------------------------------------------------------------------------
extern "C" void kernel_launch(void* const* d_in, const int* in_sizes, int n_in,
                              void* d_out, int out_size, void* d_ws, size_t ws_size,
                              hipStream_t stream)
{
  const float* Xh = (const float*)d_in[0];
  const float* Xm = (const float*)d_in[1];
  const int*   lengths = (const int*)d_in[2];
  const float* Wh = (const float*)d_in[3];
  const float* bh = (const float*)d_in[4];
  const float* Wm = (const float*)d_in[5];
  const float* bm = (const float*)d_in[6];
  const float* V  = (const float*)d_in[7];

  const long S = (long)B_ * N_ * N_;           // 8,388,608
  float* out    = (float*)d_out;
  float* scores = out;                         // [B,N,N]
  float* logZ   = out + S;                     // [B]
  float* Y      = out + S + B_;                // [B,N,N]  (also the A buffer)
  float* entr   = out + 2 * S + B_;            // [B]

  const long PROJ = (long)B_ * N_ * DH;        // 2,097,152
  float* w = (float*)d_ws;
  float* H0   = w;              w += PROJ;
  float* M0   = w;              w += PROJ;
  float* H    = w;              w += PROJ;
  float* Mm   = w;              w += PROJ;
  float* T1   = w;              w += PROJ;     // HV / Y@M
  float* T2   = w;              w += PROJ;     // Y^T@H
  float* Lhat = w;              w += (long)B_ * NM1 * LDM;
  float* Binv = w;              w += (long)B_ * NM1 * LDM;
  float* csum = w;              w += (long)B_ * N_;
  float* cmx  = w;              w += B_;
  float* ldet = w;              w += B_;
  float* dotY = w;              w += B_;

  const long sN2 = (long)N_ * N_;              // per-batch scores/Y stride
  const long sNH = (long)N_ * DH;              // per-batch H/M stride

  const dim3 gProj((B_ * N_) / 128, DH / 64, 1);   // flattened [16384 x 128]
  const dim3 gFeed(N_ / 128, DH / 64, B_);         // batched  [512 x 128]
  const dim3 gScore(N_ / 128, N_ / 64, B_);        // batched  [512 x 512]

  // ---- Projections: H0 = Xh Wh^T + bh ; H = relu(H0) (and same for M) ----
  gemm_wmma<0, 1, true, false, true, true><<<gProj, 256, 0, stream>>>(
      Xh, Wh, nullptr, bh, H, H0, DIN, DIN, DIN, DH, 0, 0, 0);
  gemm_wmma<0, 1, true, false, true, true><<<gProj, 256, 0, stream>>>(
      Xm, Wm, nullptr, bm, Mm, M0, DIN, DIN, DIN, DH, 0, 0, 0);

  for (int it = 0; it < NITER; ++it) {
    if (it > 0) {
      // T1 = Y @ M        [512x512]x[512x128], batched
      gemm_wmma<0, 0, false, false, false, false><<<gFeed, 256, 0, stream>>>(
          Y, Mm, nullptr, nullptr, T1, nullptr, N_, N_, DH, DH, sN2, sNH, sNH);
      // T2 = Y^T @ H
      gemm_wmma<1, 0, false, false, false, false><<<gFeed, 256, 0, stream>>>(
          Y, H, nullptr, nullptr, T2, nullptr, N_, N_, DH, DH, sN2, sNH, sNH);
      // H = relu(H0 + T1 @ V^T)   (flattened over batch)
      gemm_wmma<0, 1, false, true, false, true><<<gProj, 256, 0, stream>>>(
          T1, V, H0, nullptr, H, nullptr, DH, DH, DH, DH, 0, 0, 0);
      // M = relu(M0 + T2 @ V)
      gemm_wmma<0, 0, false, true, false, true><<<gProj, 256, 0, stream>>>(
          T2, V, M0, nullptr, Mm, nullptr, DH, DH, DH, DH, 0, 0, 0);
    }
    // T1 = H @ V
    gemm_wmma<0, 0, false, false, false, false><<<gProj, 256, 0, stream>>>(
        H, V, nullptr, nullptr, T1, nullptr, DH, DH, DH, DH, 0, 0, 0);
    // scores = T1 @ M^T   (batched, [512x128]x[512x128]^T)
    gemm_wmma<0, 1, false, false, false, false><<<gScore, 256, 0, stream>>>(
        T1, Mm, nullptr, nullptr, scores, nullptr, DH, DH, DH, N_, sNH, sNH, sN2);

    // ---- matrix-tree marginals ----
    cmax_kernel<<<B_, 1024, 0, stream>>>(scores, lengths, cmx);
    expA_kernel<<<2048, 1024, 0, stream>>>(scores, lengths, cmx, Y);  // A in Y region
    colsum_kernel<<<B_, 512, 0, stream>>>(Y, csum);
    lhat_kernel<<<dim3(512, 1, B_), 256, 0, stream>>>(Y, csum, lengths, Lhat);
    lu_kernel<<<B_, 1024, 0, stream>>>(Lhat, ldet);
    inv_kernel<<<B_, 512, 0, stream>>>(Lhat, Binv);
    marg_kernel<<<B_, 1024, 0, stream>>>(Y, scores, Binv, dotY);      // Y in place
    finalize_kernel<<<1, 32, 0, stream>>>(ldet, cmx, lengths, dotY, logZ, entr);
  }
}